// FakeNewsClassifier_9586367005057
// MI455X (gfx1250) — compile-verified
//
#include <hip/hip_runtime.h>

#define BDIM 8192
#define DDIM 1024
#define EDIM 8
#define OUTD 2
#define TILE_M 32     // 2 x 16-row WMMA subtiles per workgroup: B frags reused 2x
#define KC 128
#define AP 132        // padded LDS pitch (floats): 528B/row -> 4-bank step, conflict-free, 16B-aligned

typedef float v2f __attribute__((ext_vector_type(2)));
typedef float v4f __attribute__((ext_vector_type(4)));
typedef float v8f __attribute__((ext_vector_type(8)));

__global__ void zero_counts_kernel(int* counts) {
    if (threadIdx.x < EDIM) counts[threadIdx.x] = 0;
}

__global__ void compact_rows_kernel(const int* __restrict__ ids,
                                    int* __restrict__ counts,
                                    int* __restrict__ rowlist) {
    int b = blockIdx.x * blockDim.x + threadIdx.x;
    if (b < BDIM) {
        int e = ids[b];
        int pos = atomicAdd(&counts[e], 1);
        rowlist[e * BDIM + pos] = b;
    }
}

// Grouped GEMM: one workgroup = 32 rows of one expert, full N=1024, fused bias+ReLU+head.
// 8 waves; wave w owns N columns [w*128, w*128+128) as 8 WMMA 16x16 tiles x 2 M-subtiles.
__global__ __launch_bounds__(256) void moe_fused_kernel(
    const float* __restrict__ X,   // [B, D]
    const float* __restrict__ W1,  // [E, D, D]  (K rows, N contiguous)
    const float* __restrict__ B1,  // [E, D]
    const float* __restrict__ W2,  // [E, D, 2]
    const float* __restrict__ B2,  // [E, 2]
    const int*   __restrict__ counts,
    const int*   __restrict__ rowlist,
    float* __restrict__ out)       // [B, 2]
{
    __shared__ __align__(16) float As[TILE_M * AP];   // 32 x 128 A chunk (padded)
    __shared__ int   rows_s[TILE_M];
    __shared__ float wp[8][TILE_M][OUTD];             // per-wave layer-2 partials

    const int e    = blockIdx.x >> 8;                 // 256 tiles per expert
    const int tile = blockIdx.x & 255;
    const int cnt  = counts[e];
    const int m0   = tile * TILE_M;
    if (m0 >= cnt) return;                            // uniform per block

    const int t    = threadIdx.x;
    const int lane = t & 31;
    const int wave = t >> 5;

    if (t < TILE_M) {
        int i = m0 + t;
        rows_s[t] = rowlist[e * BDIM + (i < cnt ? i : cnt - 1)];
    }
    __syncthreads();

    const float* Wb = W1 + (size_t)e * DDIM * DDIM;
    const int m     = lane & 15;                      // A row within subtile / B column
    const int koff  = (lane < 16) ? 0 : 2;            // K-pair select per ISA frag layout

    v8f acc[2][8];
    const v8f vzero = {0.f, 0.f, 0.f, 0.f, 0.f, 0.f, 0.f, 0.f};
#pragma unroll
    for (int s = 0; s < 2; ++s)
#pragma unroll
        for (int i = 0; i < 8; ++i) acc[s][i] = vzero;

    for (int kc = 0; kc < DDIM; kc += KC) {
        // Stage A chunk (32 rows x 128 K) -> LDS, 16B vector, coalesced
#pragma unroll
        for (int j = 0; j < (TILE_M * KC) / (256 * 4); ++j) {   // 4 iters
            int idx = t + j * 256;
            int row = idx >> 5;                      // 32 float4 per row
            int c4  = (idx & 31) << 2;
            *(v4f*)&As[row * AP + c4] =
                *(const v4f*)&X[(size_t)rows_s[row] * DDIM + kc + c4];
        }
        __syncthreads();

#pragma unroll 2
        for (int kk = 0; kk < KC; kk += 4) {
            v2f a0 = *(const v2f*)&As[m * AP + kk + koff];
            v2f a1 = *(const v2f*)&As[(m + 16) * AP + kk + koff];
            const float* w0 = Wb + (size_t)(kc + kk + koff) * DDIM + wave * 128 + m;
#pragma unroll
            for (int nt = 0; nt < 8; ++nt) {
                v2f bf;
                bf.x = w0[nt * 16];                  // B[krow  ][n]
                bf.y = w0[nt * 16 + DDIM];           // B[krow+1][n]
                acc[0][nt] = __builtin_amdgcn_wmma_f32_16x16x4_f32(
                    false, a0, false, bf, (short)0, acc[0][nt], false, false);
                acc[1][nt] = __builtin_amdgcn_wmma_f32_16x16x4_f32(
                    false, a1, false, bf, (short)0, acc[1][nt], false, false);
            }
        }
        __syncthreads();
    }

    // ---- Epilogue: bias + ReLU + tiny [D,2] head, deterministic reductions ----
    float p[2][2][8];                                 // [subtile][out][component]
#pragma unroll
    for (int s = 0; s < 2; ++s)
#pragma unroll
        for (int o = 0; o < 2; ++o)
#pragma unroll
            for (int c = 0; c < 8; ++c) p[s][o][c] = 0.f;

#pragma unroll
    for (int nt = 0; nt < 8; ++nt) {
        int n = wave * 128 + nt * 16 + m;             // this lane's column
        float b1v = B1[e * DDIM + n];
        float w2a = W2[((size_t)e * DDIM + n) * OUTD + 0];
        float w2b = W2[((size_t)e * DDIM + n) * OUTD + 1];
#pragma unroll
        for (int s = 0; s < 2; ++s)
#pragma unroll
            for (int c = 0; c < 8; ++c) {             // VGPR c -> row c (+8 for lanes>=16)
                float h = acc[s][nt][c] + b1v;
                h = h > 0.f ? h : 0.f;
                p[s][0][c] += h * w2a;
                p[s][1][c] += h * w2b;
            }
    }
    // Reduce over the 16 lanes of each half-wave (halves hold disjoint row groups)
#pragma unroll
    for (int off = 1; off < 16; off <<= 1) {
#pragma unroll
        for (int s = 0; s < 2; ++s)
#pragma unroll
            for (int c = 0; c < 8; ++c) {
                p[s][0][c] += __shfl_xor(p[s][0][c], off, 32);
                p[s][1][c] += __shfl_xor(p[s][1][c], off, 32);
            }
    }
    if ((lane & 15) == 0) {
        int rb = (lane >> 4) * 8;                     // lane 0 -> rows 0-7, lane 16 -> rows 8-15
#pragma unroll
        for (int s = 0; s < 2; ++s)
#pragma unroll
            for (int c = 0; c < 8; ++c) {
                wp[wave][s * 16 + rb + c][0] = p[s][0][c];
                wp[wave][s * 16 + rb + c][1] = p[s][1][c];
            }
    }
    __syncthreads();

    if (t < 64) {
        int mm = t & 31, o = t >> 5;
        float s = B2[e * OUTD + o];
#pragma unroll
        for (int w = 0; w < 8; ++w) s += wp[w][mm][o];
        if (m0 + mm < cnt) out[(size_t)rows_s[mm] * OUTD + o] = s;
    }
}

extern "C" void kernel_launch(void* const* d_in, const int* in_sizes, int n_in,
                              void* d_out, int out_size, void* d_ws, size_t ws_size,
                              hipStream_t stream) {
    const float* X   = (const float*)d_in[0];   // feature   [B, D]
    const int*   ids = (const int*)  d_in[1];   // domain_ids[B]
    const float* W1  = (const float*)d_in[2];   // [E, D, D]
    const float* B1  = (const float*)d_in[3];   // [E, D]
    const float* W2  = (const float*)d_in[4];   // [E, D, 2]
    const float* B2  = (const float*)d_in[5];   // [E, 2]
    float* out = (float*)d_out;

    int* counts  = (int*)d_ws;                  // 8 counters (padded to 32)
    int* rowlist = counts + 32;                 // [E, B] compacted row indices

    zero_counts_kernel<<<1, 64, 0, stream>>>(counts);
    compact_rows_kernel<<<BDIM / 256, 256, 0, stream>>>(ids, counts, rowlist);
    moe_fused_kernel<<<EDIM * (BDIM / TILE_M), 256, 0, stream>>>(
        X, W1, B1, W2, B2, counts, rowlist, out);
}